// LinOSSLayer_5007931867599
// MI455X (gfx1250) — compile-verified
//
#include <hip/hip_runtime.h>
#include <hip/hip_bf16.h>
#include <math.h>

// Problem constants (LinOSS layer)
#define BB   16
#define SS   4096
#define DD   256
#define NN   64          // oscillators
#define HH   512
#define KK2  128         // 2*NN
#define DT_C 0.01f
#define EPS_C 1e-5f

// Scan chunking
#define CHUNK_L 128
#define CHUNK_C (SS / CHUNK_L)     // 32
#define MROWS   (BB * SS)          // 65536

// K5 A-tile LDS row stride: 132 ≡ 4 (mod 64 banks) -> the 32 lanes of a
// ds_load_b64 cover all 64 banks exactly once (half0 banks {4*lo+k+c},
// half1 {4*lo+k+2+c} are disjoint).
#define A_STRIDE 132

typedef float v2f __attribute__((ext_vector_type(2)));
typedef float v8f __attribute__((ext_vector_type(8)));
typedef unsigned int u32x4 __attribute__((ext_vector_type(4)));
typedef int i32x4 __attribute__((ext_vector_type(4)));
typedef int i32x8 __attribute__((ext_vector_type(8)));

#if defined(__has_builtin)
#if __has_builtin(__builtin_amdgcn_tensor_load_to_lds) && \
    __has_builtin(__builtin_amdgcn_s_wait_tensorcnt)
#define HAS_TDM 1
#endif
#endif

// -----------------------------------------------------------------------------
// K0: one-time pack of W_out(128x512) into pair-interleaved layout:
//   Wp[((k>>1)*H + h)*2 + (k&1)] = W_out[k*H + h]
// so a K5 B-fragment (rows k, k+1 at column h) is one global_load_b64.
// -----------------------------------------------------------------------------
__global__ __launch_bounds__(256) void k0_pack_wout(
    const float* __restrict__ Wout, float* __restrict__ Wp)
{
    const int i = blockIdx.x * 256 + threadIdx.x;   // < KK2*HH
    const int k = i >> 9;                           // / HH
    const int h = i & (HH - 1);
    Wp[((((unsigned)k >> 1) * HH + h) << 1) | (k & 1)] = Wout[i];
}

// -----------------------------------------------------------------------------
// K1: forcing = x(65536x256) @ W_in(256x64) + b_in   (fp32 WMMA 16x16x4)
// 8 waves/block, 16x64 strip per wave, 128 rows per block.
// W_in staged in LDS pair-interleaved + XOR-32 swizzle:
//   idx(k,n) = (k>>1)*128 + ((2n + (k&1)) ^ (((k>>1)&1)<<5))
// B fragment = single ds_load_b64; wave halves (pair, pair+1) hit banks
// {0..31} vs {32..63} -> conflict-free.
// -----------------------------------------------------------------------------
__global__ __launch_bounds__(256) void k1_forcing_gemm(
    const float* __restrict__ x, const float* __restrict__ Win,
    const float* __restrict__ bin, float* __restrict__ forcing)
{
    __shared__ float sWp[DD * NN];   // 64 KB, pair-interleaved
    for (int i = threadIdx.x; i < DD * NN; i += 256) {
        const int k = i >> 6, n = i & (NN - 1);
        const int pair = k >> 1, j = k & 1;
        sWp[pair * 128 + (((n << 1) | j) ^ ((pair & 1) << 5))] = Win[i];
    }
    __syncthreads();

    const int wave = threadIdx.x >> 5;
    const int lane = threadIdx.x & 31;
    const int lo = lane & 15;
    const int hi = lane >> 4;
    const int row0 = blockIdx.x * 128 + wave * 16;

    const float* arow = x + (size_t)(row0 + lo) * DD;

    v8f acc[4] = {};   // 4 N-tiles of 16x16

    for (int k = 0; k < DD; k += 4) {
        // A fragment: A[M=lo][K = k + 2*hi + j], j=0,1 (one b64 load)
        const float2 ap = *(const float2*)(arow + k + 2 * hi);
        v2f a; a.x = ap.x; a.y = ap.y;
        const int pair = (k + 2 * hi) >> 1;
        const float* wrow = &sWp[pair * 128];
        const int sw = (pair & 1) << 5;
#pragma unroll
        for (int t = 0; t < 4; ++t) {
            const float2 bp =
                *(const float2*)&wrow[(((t * 16 + lo) << 1) ^ sw)];
            v2f b; b.x = bp.x; b.y = bp.y;
            acc[t] = __builtin_amdgcn_wmma_f32_16x16x4_f32(
                false, a, false, b, (short)0, acc[t], false, false);
        }
    }

    // bias + store: C/D lane holds row = r + 8*hi, col = lo (per N-tile)
#pragma unroll
    for (int t = 0; t < 4; ++t) {
        const int col = t * 16 + lo;
        const float bv = bin[col];
#pragma unroll
        for (int r = 0; r < 8; ++r) {
            const int row = row0 + r + 8 * hi;
            forcing[(size_t)row * NN + col] = acc[t][r] + bv;
        }
    }
}

// -----------------------------------------------------------------------------
// K2: chunk-local scans, TWO oscillators per thread (2x ILP on the serial
// FMA chain, b64 forcing loads). id = (c*B + b)*(N/2) + np, lane-contiguous
// in np -> coalesced.
// -----------------------------------------------------------------------------
__global__ __launch_bounds__(256) void k2_scan_local(
    const float* __restrict__ forcing, const float* __restrict__ logw,
    const float* __restrict__ logz, float4* __restrict__ e4)
{
    const int id = blockIdx.x * 256 + threadIdx.x;   // < B*(N/2)*C = 16384
    const int np = id & 31;
    const int b = (id >> 5) & (BB - 1);
    const int c = id >> 9;
    const int n0 = np << 1;

    const float w0 = __expf(logw[n0]),     z0 = __expf(logz[n0]);
    const float w1 = __expf(logw[n0 + 1]), z1 = __expf(logz[n0 + 1]);
    const float a10_0 = -w0 * w0 * DT_C, a11_0 = 1.0f - 2.0f * z0 * w0 * DT_C;
    const float a10_1 = -w1 * w1 * DT_C, a11_1 = 1.0f - 2.0f * z1 * w1 * DT_C;

    const float2* f =
        (const float2*)(forcing + (size_t)(b * SS + c * CHUNK_L) * NN) + np;
    float sx0 = 0.f, sv0 = 0.f, sx1 = 0.f, sv1 = 0.f;
    for (int i = 0; i < CHUNK_L; ++i) {
        const float2 ft = f[(size_t)i * (NN / 2)];
        const float nx0 = fmaf(DT_C, sv0, sx0);
        const float nv0 = fmaf(a10_0, sx0, fmaf(a11_0, sv0, DT_C * ft.x));
        const float nx1 = fmaf(DT_C, sv1, sx1);
        const float nv1 = fmaf(a10_1, sx1, fmaf(a11_1, sv1, DT_C * ft.y));
        sx0 = nx0; sv0 = nv0; sx1 = nx1; sv1 = nv1;
    }
    e4[id] = make_float4(sx0, sv0, sx1, sv1);
}

// -----------------------------------------------------------------------------
// K3: chunk-boundary prefix with A^128 (7 squarings). 512 threads.
// -----------------------------------------------------------------------------
__global__ __launch_bounds__(256) void k3_scan_combine(
    const float* __restrict__ logw, const float* __restrict__ logz,
    const float4* __restrict__ e4, float4* __restrict__ sin4)
{
    const int id = blockIdx.x * 256 + threadIdx.x;   // < B*N/2 = 512
    const int np = id & 31;
    const int b = id >> 5;
    const int n0 = np << 1;

    float m[2][4];
#pragma unroll
    for (int j = 0; j < 2; ++j) {
        const float w = __expf(logw[n0 + j]);
        const float z = __expf(logz[n0 + j]);
        float m00 = 1.0f, m01 = DT_C;
        float m10 = -w * w * DT_C, m11 = 1.0f - 2.0f * z * w * DT_C;
        for (int i = 0; i < 7; ++i) {   // A^(2^7)
            const float t00 = m00 * m00 + m01 * m10;
            const float t01 = m00 * m01 + m01 * m11;
            const float t10 = m10 * m00 + m11 * m10;
            const float t11 = m10 * m01 + m11 * m11;
            m00 = t00; m01 = t01; m10 = t10; m11 = t11;
        }
        m[j][0] = m00; m[j][1] = m01; m[j][2] = m10; m[j][3] = m11;
    }

    float sx0 = 0.f, sv0 = 0.f, sx1 = 0.f, sv1 = 0.f;
    for (int c = 0; c < CHUNK_C; ++c) {
        const int idx = (c * BB + b) * (NN / 2) + np;
        sin4[idx] = make_float4(sx0, sv0, sx1, sv1);
        const float4 ec = e4[idx];
        const float nx0 = fmaf(m[0][0], sx0, fmaf(m[0][1], sv0, ec.x));
        const float nv0 = fmaf(m[0][2], sx0, fmaf(m[0][3], sv0, ec.y));
        const float nx1 = fmaf(m[1][0], sx1, fmaf(m[1][1], sv1, ec.z));
        const float nv1 = fmaf(m[1][2], sx1, fmaf(m[1][3], sv1, ec.w));
        sx0 = nx0; sv0 = nv0; sx1 = nx1; sv1 = nv1;
    }
}

// -----------------------------------------------------------------------------
// K4: final scan from corrected initial states; two oscillators/thread so
// each step stores one float4 (states [4np..4np+3] of flat(B,S,2N)).
// -----------------------------------------------------------------------------
__global__ __launch_bounds__(256) void k4_scan_final(
    const float* __restrict__ forcing, const float* __restrict__ logw,
    const float* __restrict__ logz, const float4* __restrict__ sin4,
    float4* __restrict__ flat4)
{
    const int id = blockIdx.x * 256 + threadIdx.x;
    const int np = id & 31;
    const int b = (id >> 5) & (BB - 1);
    const int c = id >> 9;
    const int n0 = np << 1;

    const float w0 = __expf(logw[n0]),     z0 = __expf(logz[n0]);
    const float w1 = __expf(logw[n0 + 1]), z1 = __expf(logz[n0 + 1]);
    const float a10_0 = -w0 * w0 * DT_C, a11_0 = 1.0f - 2.0f * z0 * w0 * DT_C;
    const float a10_1 = -w1 * w1 * DT_C, a11_1 = 1.0f - 2.0f * z1 * w1 * DT_C;

    const float4 s0 = sin4[id];
    float sx0 = s0.x, sv0 = s0.y, sx1 = s0.z, sv1 = s0.w;

    const float2* f =
        (const float2*)(forcing + (size_t)(b * SS + c * CHUNK_L) * NN) + np;
    float4* o = flat4 + (size_t)(b * SS + c * CHUNK_L) * (KK2 / 4) + np;
    for (int i = 0; i < CHUNK_L; ++i) {
        const float2 ft = f[(size_t)i * (NN / 2)];
        const float nx0 = fmaf(DT_C, sv0, sx0);
        const float nv0 = fmaf(a10_0, sx0, fmaf(a11_0, sv0, DT_C * ft.x));
        const float nx1 = fmaf(DT_C, sv1, sx1);
        const float nv1 = fmaf(a10_1, sx1, fmaf(a11_1, sv1, DT_C * ft.y));
        sx0 = nx0; sv0 = nv0; sx1 = nx1; sv1 = nv1;
        o[(size_t)i * (KK2 / 4)] = make_float4(sx0, sv0, sx1, sv1);
    }
}

// -----------------------------------------------------------------------------
// K5: out = flat(65536x128) @ W_out(128x512) + b_out, fused LayerNorm.
// Workgroup (8 waves) owns 16 rows x full 512 cols.
// A-tile (16x128) DMA'd into LDS by the Tensor Data Mover (wave 0 issues
// tensor_load_to_lds with a 2-D D#; TDM pad feature inserts 4 DWORDs every
// 128 -> 132-float rows, bank-conflict-free ds_load_b64 A-fragments).
// B-fragments are single b64 loads from the pre-packed Wp.
// Row mean/var via ds_add_f32 LDS atomics; normalize in-register.
// -----------------------------------------------------------------------------
__global__ __launch_bounds__(256) void k5_out_gemm_ln(
    const float* __restrict__ flat, const float* __restrict__ Wp,
    const float* __restrict__ bout, const float* __restrict__ gamma,
    const float* __restrict__ beta, float* __restrict__ out)
{
    __shared__ float sA[16 * A_STRIDE];   // 8.25 KB (132-float row stride)
    __shared__ float ssum[16];
    __shared__ float ssq[16];

    const int row0 = blockIdx.x * 16;
    if (threadIdx.x < 16) { ssum[threadIdx.x] = 0.0f; ssq[threadIdx.x] = 0.0f; }

#if defined(HAS_TDM)
    if (threadIdx.x < 32) {   // wave 0 drives the TDM; EXEC ignored by TDM
        const unsigned long long gaddr =
            (unsigned long long)(uintptr_t)(flat + (size_t)row0 * KK2);
        const unsigned int ldsb = (unsigned int)(uintptr_t)(&sA[0]);
        u32x4 g0;
        g0[0] = 1u;                                   // count = 1 (valid D#)
        g0[1] = ldsb;                                 // lds_addr
        g0[2] = (unsigned int)(gaddr & 0xffffffffu);  // global_addr[31:0]
        g0[3] = (unsigned int)((gaddr >> 32) & 0x01ffffffu) | (2u << 30); // type=2
        i32x8 g1;
        // data_size=4B (2), pad_enable, pad_interval=128 DW (6), pad_amount=4 DW (3)
        g1[0] = (int)((2u << 16) | (1u << 20) | (6u << 22) | (3u << 25));
        g1[1] = (int)((unsigned)KK2 << 16);   // tensor_dim0[15:0] @ bits 63:48
        g1[2] = (int)(16u << 16);             // tensor_dim1[15:0] = 16 rows
        g1[3] = (int)((unsigned)KK2 << 16);   // tile_dim0 = 128
        g1[4] = 16;                           // tile_dim1 = 16, tile_dim2 = 0
        g1[5] = KK2;                          // tensor_dim0_stride[31:0] = 128
        g1[6] = 0;
        g1[7] = 0;
        const i32x4 z4 = {0, 0, 0, 0};
#if __clang_major__ >= 23
        const i32x8 z8 = {0, 0, 0, 0, 0, 0, 0, 0};
        __builtin_amdgcn_tensor_load_to_lds(g0, g1, z4, z4, z8, 0);
#else
        __builtin_amdgcn_tensor_load_to_lds(g0, g1, z4, z4, 0);
#endif
        __builtin_amdgcn_s_wait_tensorcnt(0);
    }
#else
    for (int i = threadIdx.x; i < 16 * KK2; i += 256) {
        const int r = i >> 7, c = i & (KK2 - 1);
        sA[r * A_STRIDE + c] = flat[(size_t)(row0 + r) * KK2 + c];
    }
#endif
    __syncthreads();

    const int wave = threadIdx.x >> 5;
    const int lane = threadIdx.x & 31;
    const int lo = lane & 15;
    const int hi = lane >> 4;

    v8f acc[4] = {};
    for (int k = 0; k < KK2; k += 4) {
        const float2 ap = *(const float2*)&sA[lo * A_STRIDE + k + 2 * hi];
        v2f a; a.x = ap.x; a.y = ap.y;
        const float* wrow = Wp + ((size_t)((k + 2 * hi) >> 1) * HH) * 2;
#pragma unroll
        for (int t = 0; t < 4; ++t) {
            const int col = (wave * 4 + t) * 16 + lo;
            const float2 bp = *(const float2*)&wrow[col << 1];
            v2f b; b.x = bp.x; b.y = bp.y;
            acc[t] = __builtin_amdgcn_wmma_f32_16x16x4_f32(
                false, a, false, b, (short)0, acc[t], false, false);
        }
    }

    // bias + per-row partial moments (lane holds rows r+8*hi, 4 cols)
    float ps[8], ps2[8];
#pragma unroll
    for (int r = 0; r < 8; ++r) { ps[r] = 0.0f; ps2[r] = 0.0f; }
#pragma unroll
    for (int t = 0; t < 4; ++t) {
        const int col = (wave * 4 + t) * 16 + lo;
        const float bv = bout[col];
#pragma unroll
        for (int r = 0; r < 8; ++r) {
            const float v = acc[t][r] + bv;
            acc[t][r] = v;
            ps[r] += v;
            ps2[r] += v * v;
        }
    }
#pragma unroll
    for (int r = 0; r < 8; ++r) {
        const int row = r + 8 * hi;
        atomicAdd(&ssum[row], ps[r]);     // ds_add_f32
        atomicAdd(&ssq[row], ps2[r]);
    }
    __syncthreads();

    // normalize + store (each output element written exactly once)
#pragma unroll
    for (int r = 0; r < 8; ++r) {
        const int row = r + 8 * hi;
        const float mean = ssum[row] * (1.0f / (float)HH);
        const float var = ssq[row] * (1.0f / (float)HH) - mean * mean;
        const float inv = rsqrtf(var + EPS_C);
#pragma unroll
        for (int t = 0; t < 4; ++t) {
            const int col = (wave * 4 + t) * 16 + lo;
            out[(size_t)(row0 + row) * HH + col] =
                (acc[t][r] - mean) * inv * gamma[col] + beta[col];
        }
    }
}

// -----------------------------------------------------------------------------
extern "C" void kernel_launch(void* const* d_in, const int* in_sizes, int n_in,
                              void* d_out, int out_size, void* d_ws, size_t ws_size,
                              hipStream_t stream)
{
    const float* x     = (const float*)d_in[0];   // (B,S,D)
    const float* W_in  = (const float*)d_in[1];   // (D,N)
    const float* b_in  = (const float*)d_in[2];   // (N,)
    const float* logw  = (const float*)d_in[3];   // (N,)
    const float* logz  = (const float*)d_in[4];   // (N,)
    const float* W_out = (const float*)d_in[5];   // (2N,H)
    const float* b_out = (const float*)d_in[6];   // (H,)
    const float* gamma = (const float*)d_in[7];   // (H,)
    const float* beta  = (const float*)d_in[8];   // (H,)
    float* out = (float*)d_out;                   // (B,S,H)

    // Workspace: forcing | flat states | packed W_out | chunk-ends | chunk-starts
    float* ws = (float*)d_ws;
    float*  forcing = ws;                                       // 4,194,304 f
    float*  flatst  = forcing + (size_t)MROWS * NN;             // 8,388,608 f
    float*  Wp      = flatst + (size_t)MROWS * KK2;             //    65,536 f
    float4* e4      = (float4*)(Wp + (size_t)KK2 * HH);         //    16,384 f4
    float4* sin4    = e4 + (size_t)BB * (NN / 2) * CHUNK_C;     //    16,384 f4

    // K0: pack W_out (one-time, 256 KB)
    k0_pack_wout<<<(KK2 * HH) / 256, 256, 0, stream>>>(W_out, Wp);

    // K1: forcing GEMM (512 blocks x 256)
    k1_forcing_gemm<<<MROWS / 128, 256, 0, stream>>>(x, W_in, b_in, forcing);

    // K2: local chunk scans (16384 threads, 2 oscillators each)
    k2_scan_local<<<(BB * (NN / 2) * CHUNK_C) / 256, 256, 0, stream>>>(
        forcing, logw, logz, e4);

    // K3: chunk-boundary prefix (512 threads)
    k3_scan_combine<<<(BB * (NN / 2)) / 256, 256, 0, stream>>>(
        logw, logz, e4, sin4);

    // K4: final scan writing all states (float4 stores)
    k4_scan_final<<<(BB * (NN / 2) * CHUNK_C) / 256, 256, 0, stream>>>(
        forcing, logw, logz, sin4, (float4*)flatst);

    // K5: output GEMM + bias + LayerNorm (4096 blocks)
    k5_out_gemm_ln<<<MROWS / 16, 256, 0, stream>>>(
        flatst, Wp, b_out, gamma, beta, out);
}